// PARAFAC_1657857376964
// MI455X (gfx1250) — compile-verified
//
#include <hip/hip_runtime.h>

typedef __attribute__((ext_vector_type(2))) float v2f;
typedef __attribute__((ext_vector_type(8))) float v8f;

constexpr int KD = 256;   // K (columns of every factor)
constexpr int DA = 200;   // rows of F3 (output M)
constexpr int DB = 200;   // rows of F4 (output N)

// Kernel 1: prod[k] = F0[i0,k] * F1[i1,k] * F2[i2,k]   (256 floats -> d_ws)
__global__ __launch_bounds__(KD) void hadamard_gather_kernel(
    const float* __restrict__ F0, const float* __restrict__ F1,
    const float* __restrict__ F2, const int* __restrict__ idx,
    float* __restrict__ prod) {
  const int k = threadIdx.x;
  const long i0 = idx[0], i1 = idx[1], i2 = idx[2];
  prod[k] = F0[i0 * KD + k] * F1[i1 * KD + k] * F2[i2 * KD + k];
}

// Kernel 2: out[a*DB+b] = sum_k (F3[a,k]*prod[k]) * F4[b,k]
// One wave32 per 16x16 output tile; K swept via V_WMMA_F32_16X16X4_F32.
__global__ __launch_bounds__(32) void kr_gemm_wmma_kernel(
    const float* __restrict__ F3, const float* __restrict__ F4,
    const float* __restrict__ prod, float* __restrict__ out) {
  const int lane = threadIdx.x;      // 0..31
  const int hi   = lane >> 4;        // 0: K pair {k,k+1}; 1: K pair {k+2,k+3}
  const int mn   = lane & 15;        // row-in-tile for A/B operand fetch

  const int tm = blockIdx.x;         // tile along M (F3 rows)
  const int tn = blockIdx.y;         // tile along N (F4 rows)

  // Branch-free edge handling: clamp address, zero the operand (EXEC must
  // remain all-ones across WMMA).
  const int rowA  = tm * 16 + mn;
  const int rowB  = tn * 16 + mn;
  const float mskA = (rowA < DA) ? 1.0f : 0.0f;
  const float mskB = (rowB < DB) ? 1.0f : 0.0f;
  const int rA = (rowA < DA) ? rowA : (DA - 1);
  const int rB = (rowB < DB) ? rowB : (DB - 1);

  const float* pa = F3 + (size_t)rA * KD + 2 * hi;  // A: 16x4 fp32 operand slice
  const float* pb = F4 + (size_t)rB * KD + 2 * hi;  // B: 4x16 fp32 (= F4^T tile)
  const float* pp = prod + 2 * hi;

  v8f acc = {};
#pragma unroll 8
  for (int k = 0; k < KD; k += 4) {
    v2f a = *(const v2f*)(pa + k);
    v2f p = *(const v2f*)(pp + k);
    v2f b = *(const v2f*)(pb + k);
    a *= p;      // fold Hadamard(prod) into A
    a *= mskA;   // zero out-of-range M rows
    b *= mskB;   // zero out-of-range N rows
    // 8 args: (neg_a, A, neg_b, B, c_mod, C, reuse_a, reuse_b)
    acc = __builtin_amdgcn_wmma_f32_16x16x4_f32(
        false, a, false, b, (short)0, acc, false, false);
  }

  // D layout: col = lane&15; row-in-tile = v + (lane>=16 ? 8 : 0)
  const int col = tn * 16 + mn;
  if (col < DB) {
#pragma unroll
    for (int v = 0; v < 8; ++v) {
      const int row = tm * 16 + v + hi * 8;
      if (row < DA) out[(size_t)row * DB + col] = acc[v];
    }
  }
}

extern "C" void kernel_launch(void* const* d_in, const int* in_sizes, int n_in,
                              void* d_out, int out_size, void* d_ws, size_t ws_size,
                              hipStream_t stream) {
  const float* F0  = (const float*)d_in[0];
  const float* F1  = (const float*)d_in[1];
  const float* F2  = (const float*)d_in[2];
  const float* F3  = (const float*)d_in[3];
  const float* F4  = (const float*)d_in[4];
  const int*   idx = (const int*)d_in[5];

  float* prod = (float*)d_ws;   // 256 floats of scratch
  float* out  = (float*)d_out;  // 40000 floats

  hadamard_gather_kernel<<<1, KD, 0, stream>>>(F0, F1, F2, idx, prod);

  dim3 grid((DA + 15) / 16, (DB + 15) / 16);  // 13 x 13 tiles
  kr_gemm_wmma_kernel<<<grid, 32, 0, stream>>>(F3, F4, prod, out);
}